// c_tildes_weight_1305670057992
// MI455X (gfx1250) — compile-verified
//
#include <hip/hip_runtime.h>
#include <stdint.h>

// ---------------------------------------------------------------------------
// MACE-style sparse contraction: three scatter-add SpMMs.
//   out1[c] += val1*w1[r]   (360 nnz   -> 360 floats)
//   out2[r] += val2*w2[c]   (2M nnz    -> 129,600 floats)
//   out3[r] += val3*w3[c]   (8M nnz    -> 46,656,000 floats, ~178MB)
// Bandwidth-bound (0.04 flop/byte). Strategy: zero-fill keeps the output
// dirty/resident in the 192MB L2 so the 10M float atomics RMW in-cache; the
// coupling-weight tables (8/32KB) are staged into LDS with the CDNA5 Tensor
// Data Mover (one tensor_load_to_lds DMA per workgroup, TENSORcnt), falling
// back to global_load_async_to_lds_b128 (ASYNCcnt) if the TDM builtin is
// unavailable; index/value streams are read as B128.
// ---------------------------------------------------------------------------

typedef unsigned int u32x4 __attribute__((ext_vector_type(4)));
typedef int          i32x8 __attribute__((ext_vector_type(8)));
typedef int          i32x4 __attribute__((ext_vector_type(4)));

#if defined(__has_builtin)
#if __has_builtin(__builtin_amdgcn_tensor_load_to_lds)
#define HAVE_TDM 1
#endif
#endif

// Stage K floats of w[] into LDS.
__device__ __forceinline__ void stage_w_to_lds(const float* __restrict__ w,
                                               float* s_w, int K) {
#ifdef HAVE_TDM
  // One wave issues a single TDM DMA for the whole table. EXEC is ignored by
  // tensor ops, but the branch is wave-uniform so only wave 0 executes it.
  if (threadIdx.x < 32) {
    const uint64_t ga  = (uint64_t)(uintptr_t)w;
    const uint32_t lds = (uint32_t)(uintptr_t)s_w;  // low 32 bits of a generic
    // shared pointer = workgroup-relative LDS byte address.
    const uint32_t k   = (uint32_t)K;

    u32x4 g0;
    g0[0] = 1u;                                   // count=1, user D#, no gather
    g0[1] = lds;                                  // lds_addr (bytes)
    g0[2] = (uint32_t)ga;                         // global_addr[31:0]
    g0[3] = (uint32_t)((ga >> 32) & 0x01FFFFFFu)  // global_addr[56:32]
          | (2u << 30);                           // type=2 ("image")

    i32x8 g1;
    g1[0] = (int)(2u << 16);                      // wg_mask=0, data_size=4B
    g1[1] = (int)((k & 0xFFFFu) << 16);           // tensor_dim0[15:0]
    g1[2] = (int)((k >> 16) | (1u << 16));        // tensor_dim0[31:16], dim1=1
    g1[3] = (int)((k & 0xFFFFu) << 16);           // tile_dim0 = K
    g1[4] = 1;                                    // tile_dim1 = 1
    g1[5] = (int)k;                               // tensor_dim0_stride[31:0]=K
    g1[6] = 0;
    g1[7] = 0;

    i32x4 gz = {0, 0, 0, 0};
#if __clang_major__ >= 23
    i32x8 gz8 = {0, 0, 0, 0, 0, 0, 0, 0};
    __builtin_amdgcn_tensor_load_to_lds(g0, g1, gz, gz, gz8, 0);
#else
    __builtin_amdgcn_tensor_load_to_lds(g0, g1, gz, gz, 0);
#endif
    __builtin_amdgcn_s_wait_tensorcnt(0);         // s_wait_tensorcnt 0
  }
#else
  // Fallback: per-thread async copies (ASYNCcnt), b128 granularity.
  const uint32_t lds_base = (uint32_t)(uintptr_t)s_w;
  const int k4n = K >> 2;
  for (int k4 = threadIdx.x; k4 < k4n; k4 += blockDim.x) {
    uint32_t goff = (uint32_t)(k4 << 4);
    uint32_t loff = lds_base + goff;
    asm volatile("global_load_async_to_lds_b128 %0, %1, %2"
                 :: "v"(loff), "v"(goff), "s"(w)
                 : "memory");
  }
  for (int kk = (k4n << 2) + threadIdx.x; kk < K; kk += blockDim.x)
    s_w[kk] = w[kk];
  asm volatile("s_wait_asynccnt 0" ::: "memory");
#endif
  __syncthreads();   // publish LDS table to all waves
}

__global__ void zero_kernel(float4* __restrict__ out4, float* __restrict__ out,
                            long n4, long n) {
  long stride = (long)gridDim.x * blockDim.x;
  for (long i = (long)blockIdx.x * blockDim.x + threadIdx.x; i < n4; i += stride)
    out4[i] = make_float4(0.f, 0.f, 0.f, 0.f);
  long gid = (long)blockIdx.x * blockDim.x + threadIdx.x;
  long tail0 = n4 << 2;
  if (tail0 + gid < n) out[tail0 + gid] = 0.f;
}

// Order 1: gather at ROW, scatter-add at COL. Tiny (360 nnz).
__global__ void order1_kernel(const int* __restrict__ row,
                              const int* __restrict__ col,
                              const float* __restrict__ val,
                              const float* __restrict__ w,
                              float* __restrict__ out, int nnz) {
  int i = blockIdx.x * blockDim.x + threadIdx.x;
  if (i < nnz) atomicAdd(&out[col[i]], val[i] * w[row[i]]);
}

// Orders 2/3: gather w at COL (LDS-resident), scatter-add at ROW.
__global__ void scatter_spmm_kernel(const int* __restrict__ row,
                                    const int* __restrict__ col,
                                    const float* __restrict__ val,
                                    const float* __restrict__ w,
                                    float* __restrict__ out,
                                    int nnz, int K) {
  extern __shared__ float s_w[];
  stage_w_to_lds(w, s_w, K);

  const int nv = nnz >> 2;
  const int4*   row4 = (const int4*)row;
  const int4*   col4 = (const int4*)col;
  const float4* val4 = (const float4*)val;
  const int stride = gridDim.x * blockDim.x;
  for (int i = blockIdx.x * blockDim.x + threadIdx.x; i < nv; i += stride) {
    // pull the next grid-stride chunk toward L2 (global_prefetch_b8)
    __builtin_prefetch(&row4[i + stride], 0, 1);
    __builtin_prefetch(&col4[i + stride], 0, 1);
    __builtin_prefetch(&val4[i + stride], 0, 1);
    int4   r = row4[i];
    int4   c = col4[i];
    float4 v = val4[i];
    atomicAdd(&out[r.x], v.x * s_w[c.x]);
    atomicAdd(&out[r.y], v.y * s_w[c.y]);
    atomicAdd(&out[r.z], v.z * s_w[c.z]);
    atomicAdd(&out[r.w], v.w * s_w[c.w]);
  }
  // tail (nnz not multiple of 4)
  int tail = nnz & 3;
  int gid = blockIdx.x * blockDim.x + threadIdx.x;
  if (gid < tail) {
    int i = nnz - tail + gid;
    atomicAdd(&out[row[i]], val[i] * s_w[col[i]]);
  }
}

extern "C" void kernel_launch(void* const* d_in, const int* in_sizes, int n_in,
                              void* d_out, int out_size, void* d_ws, size_t ws_size,
                              hipStream_t stream) {
  // setup_inputs() order:
  //  0:row1 1:col1 2:val1 3:w1  4:row2 5:col2 6:val2 7:w2  8:row3 9:col3 10:val3 11:w3
  const int*   row1 = (const int*)d_in[0];
  const int*   col1 = (const int*)d_in[1];
  const float* val1 = (const float*)d_in[2];
  const float* w1   = (const float*)d_in[3];
  const int*   row2 = (const int*)d_in[4];
  const int*   col2 = (const int*)d_in[5];
  const float* val2 = (const float*)d_in[6];
  const float* w2   = (const float*)d_in[7];
  const int*   row3 = (const int*)d_in[8];
  const int*   col3 = (const int*)d_in[9];
  const float* val3 = (const float*)d_in[10];
  const float* w3   = (const float*)d_in[11];

  const int NNZ1 = in_sizes[0];
  const int NNZ2 = in_sizes[4];
  const int NNZ3 = in_sizes[8];
  const int D    = in_sizes[3];   // w1 is (D,1)
  const int K2   = in_sizes[7];
  const int K3   = in_sizes[11];
  const long D2  = (long)D * D;

  float* out = (float*)d_out;

  // 1) Zero the full output; RT stores keep it dirty/resident in the 192MB L2
  //    so the scatter atomics below RMW in-cache.
  long n  = (long)out_size;
  long n4 = n >> 2;
  zero_kernel<<<4096, 256, 0, stream>>>((float4*)out, out, n4, n);

  // 2) Order 1 (tiny).
  order1_kernel<<<(NNZ1 + 255) / 256, 256, 0, stream>>>(
      row1, col1, val1, w1, out, NNZ1);

  // 3) Order 2: 2M nnz, 8KB weight table staged by TDM into LDS.
  scatter_spmm_kernel<<<1024, 256, (size_t)K2 * sizeof(float), stream>>>(
      row2, col2, val2, w2, out + D, NNZ2, K2);

  // 4) Order 3: 8M nnz, 32KB weight table (320KB/WGP -> ~10 blocks/WGP).
  scatter_spmm_kernel<<<2048, 256, (size_t)K3 * sizeof(float), stream>>>(
      row3, col3, val3, w3, out + D + D2, NNZ3, K3);
}